// JeffressLinear_73452530696744
// MI455X (gfx1250) — compile-verified
//
#include <hip/hip_runtime.h>
#include <hip/hip_bf16.h>

#define T_DIM 64
#define N_DIM 64
#define C_DIM 128
#define D_OUT 31
#define CB    64                    // channels per block (C split across gridDim.y)
#define ROWW  (CB * 2 + 4)          // LDS row stride in words: padded, rows stay 16B-aligned
#define ROWB  (ROWW * 4)            // row stride in bytes (528)
#define TPLANE (N_DIM * C_DIM * 2)      // words per t-plane of x  (16384)
#define OPLANE (N_DIM * C_DIM * D_OUT)  // words per t-plane of out (253952)

__global__ __launch_bounds__(256) void jeffress_lif_kernel(
    const float* __restrict__ x,
    const float* __restrict__ delay_latent,
    const float* __restrict__ weight,
    float* __restrict__ out)
{
    __shared__ __align__(16) float tile[T_DIM * ROWW];   // ~33.8 KB
    __shared__ int s_amax[CB * 2];
    __shared__ int s_d0[D_OUT], s_d1[D_OUT];

    const int tid = threadIdx.x;
    const int n   = blockIdx.x;
    const int ch  = blockIdx.y;

    // Low 32 bits of a flat LDS pointer are the wave-relative LDS byte offset
    // (CDNA5 aperture mapping), which is what the async-to-LDS VDST expects.
    const unsigned lds_base = (unsigned)(size_t)(void*)tile;

    // ---- Stage x[t, n, ch*CB .. ch*CB+CB-1, :] into LDS via async DMA (ASYNCcnt) ----
    const float* xsrc = x + (size_t)n * (C_DIM * 2) + (size_t)ch * (CB * 2);
    for (int q = tid; q < T_DIM * (CB * 2 / 4); q += 256) {   // 64 t-rows x 32 B128 chunks
        const int t = q >> 5;
        const int k = q & 31;
        const float* src = xsrc + (size_t)t * TPLANE + k * 4;
        const unsigned dst = lds_base + (unsigned)(t * ROWB + k * 16);
        asm volatile("global_load_async_to_lds_b128 %0, %1, off"
                     :: "v"(dst), "v"(src) : "memory");
    }

    // Integer delays from the latent (independent of the tile; overlap with DMA).
    if (tid < D_OUT) {
        const float dl = delay_latent[tid];
        s_d0[tid] = (int)floorf(fmaxf(dl, 0.0f));    // relu(d)  column 0
        s_d1[tid] = (int)floorf(fmaxf(-dl, 0.0f));   // relu(-d) column 1
    }

    asm volatile("s_wait_asynccnt 0" ::: "memory");
    __syncthreads();

    // ---- argmax over t per (c, i): first occurrence of the max (strict >) ----
    if (tid < CB * 2) {
        float best = tile[tid];
        int bi = 0;
        for (int t = 1; t < T_DIM; ++t) {
            const float v = tile[t * ROWW + tid];
            if (v > best) { best = v; bi = t; }
        }
        s_amax[tid] = bi;
    }
    __syncthreads();

    const float w = weight[0];

    // ---- LIF over the time-shifted views; u = c*31 + j so wave stores are contiguous ----
    for (int u = tid; u < CB * D_OUT; u += 256) {
        const int cl = u / D_OUT;
        const int j  = u - cl * D_OUT;
        const int c  = ch * CB + cl;

        const int d0 = min(s_d0[j], (T_DIM - 1) - s_amax[2 * cl]);
        const int d1 = min(s_d1[j], (T_DIM - 1) - s_amax[2 * cl + 1]);
        int t0 = (T_DIM - d0) & (T_DIM - 1);   // source index for t = 0: (0 - d) mod 64
        int t1 = (T_DIM - d1) & (T_DIM - 1);

        float v0 = 0.0f, v1 = 0.0f;
        const int cc = cl * 2;
        float* op = out + ((size_t)n * C_DIM + c) * D_OUT + j;

        #pragma unroll 4
        for (int t = 0; t < T_DIM; ++t) {
            const float x0 = tile[t0 * ROWW + cc];
            const float x1 = tile[t1 * ROWW + cc + 1];
            v0 += (x0 - v0) * 0.5f;            // exact order of v + (x - v)/2
            v1 += (x1 - v1) * 0.5f;
            float s = 0.0f;
            if (v0 >= 1.0f) { s += 1.0f; v0 = 0.0f; }   // heaviside + hard reset
            if (v1 >= 1.0f) { s += 1.0f; v1 = 0.0f; }
            __builtin_nontemporal_store(s * w, op);     // write-once 65 MB stream
            op += OPLANE;
            t0 = (t0 + 1) & (T_DIM - 1);
            t1 = (t1 + 1) & (T_DIM - 1);
        }
    }
}

extern "C" void kernel_launch(void* const* d_in, const int* in_sizes, int n_in,
                              void* d_out, int out_size, void* d_ws, size_t ws_size,
                              hipStream_t stream) {
    const float* x  = (const float*)d_in[0];   // (64, 64, 128, 2) f32
    const float* dl = (const float*)d_in[1];   // (31, 1) f32
    const float* w  = (const float*)d_in[2];   // scalar f32
    float* out = (float*)d_out;                // (64, 64, 128, 31) f32

    dim3 grid(N_DIM, C_DIM / CB);              // 64 x 2 = 128 blocks
    jeffress_lif_kernel<<<grid, 256, 0, stream>>>(x, dl, w, out);
}